// PraxisEncoder_14525579395261
// MI455X (gfx1250) — compile-verified
//
#include <hip/hip_runtime.h>
#include <math.h>

// Segmented top-2-mean pooling (bs=8, seq=4096, emb=512, P=1024, k=2).
// HBM-bound streaming reduction: 64MB read + 16MB write @ 23.3 TB/s ~= 3.4us
// floor. Each h row is read exactly once via coalesced global_load_b128;
// per-(b,p) segments located with a uniform (scalarized) binary search over
// the sorted patch_ids (128KB, resident in 192MB L2). No WMMA: max-reduction
// has zero dot-product structure; no LDS staging: zero data reuse, so
// global->VGPR streaming is the optimal CDNA5 path. gfx1250 global_prefetch
// keeps the next token row in flight on the short (avg ~4 row) segments.

#define NEG_BIG (-1.0e9f)

template <int P, int EMB>
__global__ __launch_bounds__(128, 8)
void topk2_mean_pool_kernel(const float* __restrict__ h,
                            const int*   __restrict__ patch_ids,
                            float*       __restrict__ out,
                            int seq_len)
{
    constexpr int V4 = EMB / 4;            // 128 float4 per token row
    static_assert((P & (P - 1)) == 0, "P must be power of two");

    const int bp = blockIdx.x;             // bp = b * P + p
    const int b  = bp >> 10;               // P == 1024
    const int p  = bp & (P - 1);

    const int* __restrict__ row = patch_ids + (size_t)b * seq_len;

    // lower_bound(p): first index with row[i] >= p   (uniform -> s_loads)
    int lo = 0, hi = seq_len;
    while (lo < hi) {
        int mid = (lo + hi) >> 1;
        if (row[mid] < p) lo = mid + 1; else hi = mid;
    }
    const int start = lo;
    // upper_bound(p): first index with row[i] > p
    hi = seq_len;
    while (lo < hi) {
        int mid = (lo + hi) >> 1;
        if (row[mid] <= p) lo = mid + 1; else hi = mid;
    }
    const int count = lo - start;

    const int tid = threadIdx.x;           // 128 threads x float4 = 512 dims
    float4* __restrict__ o = reinterpret_cast<float4*>(out) + (size_t)bp * V4;

    if (count == 0) {
        o[tid] = make_float4(0.f, 0.f, 0.f, 0.f);
        return;
    }

    const float4* __restrict__ base =
        reinterpret_cast<const float4*>(h + (size_t)b * seq_len * EMB)
        + (size_t)start * V4 + tid;

    float4 m1 = make_float4(-INFINITY, -INFINITY, -INFINITY, -INFINITY);
    float4 m2 = m1;

    // 2-deep pipeline: next row's global_load_b128 issues before the VALU
    // top-2 update of the current row; prefetch covers row t+2.
    float4 v = base[0];
    for (int t = 0; t < count; ++t) {
        float4 vn;
        if (t + 1 < count) {
            vn = base[(size_t)(t + 1) * V4];
            if (t + 2 < count)
                __builtin_prefetch(base + (size_t)(t + 2) * V4, 0, 1); // global_prefetch_b8
        }
        // Tie-exact running top-2 per component:
        //   v > m1 : demote m1 to m2, take v
        //   v < m1 : candidate for m2
        //   v == m1: duplicate of running max -> excluded (reference masks
        //            ALL tokens equal to the segment max before 2nd amax)
#define UPD(c)                                                \
        if (v.c > m1.c)      { m2.c = m1.c; m1.c = v.c; }     \
        else if (v.c < m1.c) { m2.c = fmaxf(m2.c, v.c); }
        UPD(x) UPD(y) UPD(z) UPD(w)
#undef UPD
        v = vn;
    }

    float4 r;
    if (count == 1) {
        r = m1;   // single token: mean of one value
    } else {
        // every segment with count>=1 has >=1 masked token contributing
        // -1e9 to the reference's 2nd amax -> max(m2, -1e9) is bit-exact
        r.x = 0.5f * (m1.x + fmaxf(m2.x, NEG_BIG));
        r.y = 0.5f * (m1.y + fmaxf(m2.y, NEG_BIG));
        r.z = 0.5f * (m1.z + fmaxf(m2.z, NEG_BIG));
        r.w = 0.5f * (m1.w + fmaxf(m2.w, NEG_BIG));
    }
    o[tid] = r;   // coalesced global_store_b128
}

extern "C" void kernel_launch(void* const* d_in, const int* in_sizes, int n_in,
                              void* d_out, int out_size, void* d_ws, size_t ws_size,
                              hipStream_t stream) {
    (void)n_in; (void)d_ws; (void)ws_size;
    const float* h   = (const float*)d_in[0];   // [bs, seq, 512] f32
    const int*   pid = (const int*)d_in[1];     // [bs, seq] i32, sorted per batch
    float*       out = (float*)d_out;           // [bs, 1024, 512] f32

    constexpr int P   = 1024;                   // max_num_patches (static)
    constexpr int EMB = 512;
    const int bseq = in_sizes[1];               // bs * seq
    const int bs   = out_size / (P * EMB);      // = 8
    const int seq  = bseq / bs;                 // = 4096

    dim3 grid(bs * P);                          // one block per (batch, patch)
    topk2_mean_pool_kernel<P, EMB><<<grid, 128, 0, stream>>>(h, pid, out, seq);
}